// GNNModel_1795296329975
// MI455X (gfx1250) — compile-verified
//
#include <hip/hip_runtime.h>

#define N_NODES   100000
#define N_EDGES   1600000
#define N_FEAT    128
#define HIDDEN    128
#define N_CLASSES 8
#define N_GRAPHS  256

typedef __attribute__((ext_vector_type(2))) float v2f;
typedef __attribute__((ext_vector_type(8))) float v8f;

// ---------------------------------------------------------------------------
// Fire-and-forget device-scope f32 atomic add (GLOBAL_ATOMIC_ADD_F32,
// non-returning form, STOREcnt-tracked). Guarantees the hardware atomic
// instead of a possible CAS loop from atomicAdd().
// ---------------------------------------------------------------------------
__device__ __forceinline__ void gatomic_fadd(float* p, float v)
{
    asm volatile("global_atomic_add_f32 %0, %1, off scope:SCOPE_DEV"
                 :: "v"(p), "v"(v) : "memory");
}

// ---------------------------------------------------------------------------
// Degree: deg[i] = 1 + in-degree(i); dinv = rsqrt(deg); deginv = 1/deg
// ---------------------------------------------------------------------------
__global__ __launch_bounds__(256) void deg_count_kernel(
    const int* __restrict__ dst, float* __restrict__ deg)
{
    int e = blockIdx.x * 256 + threadIdx.x;
    if (e < N_EDGES) gatomic_fadd(&deg[dst[e]], 1.0f);
}

__global__ __launch_bounds__(256) void deg_finalize_kernel(
    const float* __restrict__ deg, float* __restrict__ dinv,
    float* __restrict__ deginv)
{
    int i = blockIdx.x * 256 + threadIdx.x;
    if (i < N_NODES) {
        float d = deg[i] + 1.0f;
        dinv[i]   = rsqrtf(d);
        deginv[i] = 1.0f / d;
    }
}

// ---------------------------------------------------------------------------
// C[M,128] = relu_lo(A)[M,128] @ W[128,128], fp32 WMMA 16x16x4.
// Block: 32 rows x 128 cols; 8 waves; wave w owns two stacked 16x16 tiles at
// column n0 = 16*w (B fragments reused for both). A tile (32x128) staged in
// LDS with pitch 132 via async global->LDS B128 copies (ASYNCcnt).
// "lo" is the activation lower bound applied on A-fragment read:
// -inf = identity, 0 = ReLU (fused activation of the previous layer).
// ---------------------------------------------------------------------------
__global__ __launch_bounds__(256) void gemm_wmma_f32_kernel(
    const float* __restrict__ A, const float* __restrict__ W,
    float* __restrict__ C, float lo)
{
    __shared__ float sA[32 * 132];          // 32 rows, pitch 132 (16.5 KB)

    const int tid = threadIdx.x;
    const int m0  = blockIdx.x * 32;

    // Async staging of the 32x128 A tile (1024 float4 / 256 threads).
    const float4* A4 = (const float4*)(A + (size_t)m0 * 128);
    #pragma unroll
    for (int i = tid; i < 1024; i += 256) {
        int r = i >> 5;                               // row 0..31
        int c = i & 31;                               // float4 col 0..31
        unsigned ldsOff = (unsigned)(size_t)&sA[r * 132 + c * 4];
        const float4* gp = &A4[i];
        asm volatile("global_load_async_to_lds_b128 %0, %1, off"
                     :: "v"(ldsOff), "v"(gp) : "memory");
    }
    asm volatile("s_wait_asynccnt 0x0" ::: "memory");
    __syncthreads();

    const int lane = tid & 31;
    const int wave = tid >> 5;
    const int half = lane >> 4;              // 0: lanes 0-15, 1: lanes 16-31
    const int l15  = lane & 15;
    const int koff = half * 2;               // A/B K-offset within step of 4
    const int n0   = wave * 16;

    v8f acc0 = {};
    v8f acc1 = {};
    #pragma unroll 4
    for (int k0 = 0; k0 < 128; k0 += 4) {
        // B 4x16 frag: N striped over lanes, K over (VGPR, lane-half)
        v2f b;
        b.x = W[(k0 + koff) * 128 + n0 + l15];
        b.y = W[(k0 + koff + 1) * 128 + n0 + l15];
        // A 16x4 frags for the two stacked M tiles (fused activation)
        v2f a0, a1;
        a0.x = fmaxf(sA[l15 * 132 + k0 + koff],            lo);
        a0.y = fmaxf(sA[l15 * 132 + k0 + koff + 1],        lo);
        a1.x = fmaxf(sA[(l15 + 16) * 132 + k0 + koff],     lo);
        a1.y = fmaxf(sA[(l15 + 16) * 132 + k0 + koff + 1], lo);
        acc0 = __builtin_amdgcn_wmma_f32_16x16x4_f32(
            false, a0, false, b, (short)0, acc0, false, false);
        acc1 = __builtin_amdgcn_wmma_f32_16x16x4_f32(
            false, a1, false, b, (short)0, acc1, false, false);
    }

    // C/D 16x16 f32: VGPR r -> M = r (+8 for lanes 16-31), N = lane&15
    const int scol = n0 + l15;
    const int r0   = m0 + half * 8;
    #pragma unroll
    for (int r = 0; r < 8; ++r)
        C[(size_t)(r0 + r) * 128 + scol] = acc0[r];
    #pragma unroll
    for (int r = 0; r < 8; ++r)
        C[(size_t)(r0 + 16 + r) * 128 + scol] = acc1[r];
}

// ---------------------------------------------------------------------------
// Q = P * deginv[node] + b[feat]   (self-loop + bias seed; replaces memset)
// float4-vectorized: one thread per 4 features.
// ---------------------------------------------------------------------------
__global__ __launch_bounds__(256) void node_init_kernel(
    float* __restrict__ Q, const float* __restrict__ P,
    const float* __restrict__ deginv, const float* __restrict__ b)
{
    int idx = blockIdx.x * 256 + threadIdx.x;       // float4 index
    if (idx >= (N_NODES * HIDDEN) / 4) return;
    int f4 = idx & 31;
    int nd = idx >> 5;
    float dg  = deginv[nd];
    float4 p  = ((const float4*)P)[idx];
    float4 bb = ((const float4*)b)[f4];
    float4 q;
    q.x = p.x * dg + bb.x;
    q.y = p.y * dg + bb.y;
    q.z = p.z * dg + bb.z;
    q.w = p.w * dg + bb.w;
    ((float4*)Q)[idx] = q;
}

// ---------------------------------------------------------------------------
// Edge aggregation: Q[dst] += P[src] * (dinv[src]*dinv[dst])
// One wave (32 lanes) per edge: 512B float4 gather + 4 hw atomics per lane.
// ---------------------------------------------------------------------------
__global__ __launch_bounds__(256) void edge_agg_kernel(
    const float* __restrict__ H, const int* __restrict__ src,
    const int* __restrict__ dst, const float* __restrict__ dinv,
    float* __restrict__ out)
{
    int e = blockIdx.x * 8 + (threadIdx.x >> 5);
    int lane = threadIdx.x & 31;
    if (e >= N_EDGES) return;
    int s = src[e];
    int d = dst[e];
    float norm = dinv[s] * dinv[d];
    float4 v = ((const float4*)(H + (size_t)s * 128))[lane];
    float* o = out + (size_t)d * 128 + lane * 4;
    gatomic_fadd(o + 0, v.x * norm);
    gatomic_fadd(o + 1, v.y * norm);
    gatomic_fadd(o + 2, v.z * norm);
    gatomic_fadd(o + 3, v.w * norm);
}

// ---------------------------------------------------------------------------
// Mean-pool accumulation with fused ReLU of layer 3:
// pooled[batch[n]] += relu(H[n]); counts[batch[n]] += 1
// ---------------------------------------------------------------------------
__global__ __launch_bounds__(256) void pool_kernel(
    const float* __restrict__ H, const int* __restrict__ batch,
    float* __restrict__ pooled, float* __restrict__ counts)
{
    size_t idx = (size_t)blockIdx.x * 256 + threadIdx.x;
    if (idx >= (size_t)N_NODES * HIDDEN) return;
    int f  = (int)(idx & 127);
    int nd = (int)(idx >> 7);
    int g  = batch[nd];
    gatomic_fadd(&pooled[(size_t)g * HIDDEN + f], fmaxf(H[idx], 0.0f));
    if (f == 0) gatomic_fadd(&counts[g], 1.0f);
}

// ---------------------------------------------------------------------------
// Classifier: out = relu(mean_pooled @ Wc1 + bc1) @ Wc2 + bc2
// One block (64 threads) per graph.
// ---------------------------------------------------------------------------
__global__ __launch_bounds__(64) void classifier_kernel(
    const float* __restrict__ pooled, const float* __restrict__ counts,
    const float* __restrict__ Wc1, const float* __restrict__ bc1,
    const float* __restrict__ Wc2, const float* __restrict__ bc2,
    float* __restrict__ out)
{
    __shared__ float z[HIDDEN / 2];
    int g = blockIdx.x;
    int j = threadIdx.x;                      // 0..63
    float inv = 1.0f / fmaxf(counts[g], 1.0f);
    float acc = bc1[j];
    const float* p = pooled + (size_t)g * HIDDEN;
    #pragma unroll 4
    for (int k = 0; k < HIDDEN; ++k)
        acc += p[k] * inv * Wc1[k * (HIDDEN / 2) + j];
    z[j] = fmaxf(acc, 0.0f);
    __syncthreads();
    if (j < N_CLASSES) {
        float o = bc2[j];
        #pragma unroll 4
        for (int k = 0; k < HIDDEN / 2; ++k)
            o += z[k] * Wc2[k * N_CLASSES + j];
        out[(size_t)g * N_CLASSES + j] = o;
    }
}

// ---------------------------------------------------------------------------
extern "C" void kernel_launch(void* const* d_in, const int* in_sizes, int n_in,
                              void* d_out, int out_size, void* d_ws, size_t ws_size,
                              hipStream_t stream)
{
    const float* x    = (const float*)d_in[0];
    const int*   ei   = (const int*)d_in[1];     // [2, N_EDGES] flattened
    const int*   batch= (const int*)d_in[2];
    const float* W1   = (const float*)d_in[3];
    const float* b1   = (const float*)d_in[4];
    const float* W2   = (const float*)d_in[5];
    const float* b2   = (const float*)d_in[6];
    const float* W3   = (const float*)d_in[7];
    const float* b3   = (const float*)d_in[8];
    const float* Wc1  = (const float*)d_in[9];
    const float* bc1  = (const float*)d_in[10];
    const float* Wc2  = (const float*)d_in[11];
    const float* bc2  = (const float*)d_in[12];
    const int* src  = ei;
    const int* dstv = ei + N_EDGES;

    // Workspace carve-out
    char* ws = (char*)d_ws;
    size_t off = 0;
    auto carve = [&](size_t bytes) -> void* {
        void* p = ws + off;
        off = (off + bytes + 255) & ~(size_t)255;
        return p;
    };
    const size_t featBytes = sizeof(float) * (size_t)N_NODES * HIDDEN;
    float* P      = (float*)carve(featBytes);                 // GEMM output
    float* Q      = (float*)carve(featBytes);                 // agg / layer out
    float* deg    = (float*)carve(sizeof(float) * N_NODES);
    float* dinv   = (float*)carve(sizeof(float) * N_NODES);
    float* deginv = (float*)carve(sizeof(float) * N_NODES);
    float* pooled = (float*)carve(sizeof(float) * N_GRAPHS * HIDDEN);
    float* counts = (float*)carve(sizeof(float) * N_GRAPHS);
    if (off > ws_size) return;   // workspace too small; nothing safe to do

    // Degree / normalization
    hipMemsetAsync(deg, 0, sizeof(float) * N_NODES, stream);
    deg_count_kernel<<<(N_EDGES + 255) / 256, 256, 0, stream>>>(dstv, deg);
    deg_finalize_kernel<<<(N_NODES + 255) / 256, 256, 0, stream>>>(deg, dinv, deginv);

    const float* Hin  = x;
    const float* Wl[3] = {W1, W2, W3};
    const float* bl[3] = {b1, b2, b3};
    const int gemmBlocks = N_NODES / 32;                            // 3125
    const int initBlocks = (N_NODES * HIDDEN / 4) / 256;            // 12500
    const int elemBlocks = (int)(((size_t)N_NODES * HIDDEN) / 256); // 50000

    for (int l = 0; l < 3; ++l) {
        float lo = (l == 0) ? -INFINITY : 0.0f;   // fused ReLU of prev layer
        gemm_wmma_f32_kernel<<<gemmBlocks, 256, 0, stream>>>(Hin, Wl[l], P, lo);
        node_init_kernel<<<initBlocks, 256, 0, stream>>>(Q, P, deginv, bl[l]);
        edge_agg_kernel<<<N_EDGES / 8, 256, 0, stream>>>(P, src, dstv, dinv, Q);
        Hin = Q;
    }

    // Pool (fused ReLU) + classifier
    hipMemsetAsync(pooled, 0, sizeof(float) * N_GRAPHS * HIDDEN, stream);
    hipMemsetAsync(counts, 0, sizeof(float) * N_GRAPHS, stream);
    pool_kernel<<<elemBlocks, 256, 0, stream>>>(Q, batch, pooled, counts);
    classifier_kernel<<<N_GRAPHS, 64, 0, stream>>>(pooled, counts, Wc1, bc1,
                                                   Wc2, bc2, (float*)d_out);
}